// conj_function_deblur_8031588844049
// MI455X (gfx1250) — compile-verified
//
#include <hip/hip_runtime.h>
#include <hip/hip_bf16.h>

typedef float v2f __attribute__((ext_vector_type(2)));
typedef float v8f __attribute__((ext_vector_type(8)));

#define NCH 3
#define IMH 2048
#define IMW 2048
#define KS  25
#define KR  12
#define PATCH 40           // 16 + 2*12
#define PSTR  42           // even stride -> 8B aligned b64 LDS loads, low conflicts
#define CWAVES 4           // waves (tiles) per block

// ---------------------------------------------------------------------------
// 25x25 circular correlation via V_WMMA_F32_16X16X4_F32.
// One wave computes one 16x16 output tile:
//   D(m=y,n=x) = sum_dy sum_kk A(m,kk)*B_dy(kk,n)
//   A(m,kk) = in[y0+dy-12+m, x0+kk-12]   (from 40x40 LDS patch, row = dy+m)
//   B_dy(kk,n) = kern[dy, kk-n]          (Toeplitz band, from padded LDS row)
// ---------------------------------------------------------------------------
__global__ __launch_bounds__(128)
void conv25_wmma(const float* __restrict__ in, const float* __restrict__ kern,
                 float* __restrict__ out, int flipk)
{
    __shared__ float kpad[KS * 64];             // row dy at dy*64, taps at +16..+40, zeros elsewhere
    __shared__ float patch[CWAVES][PATCH * PSTR];

    const int tid  = threadIdx.x;
    const int lane = tid & 31;
    const int wave = tid >> 5;
    const int half = (lane >> 4) & 1;           // 0: lanes 0-15, 1: lanes 16-31
    const int l15  = lane & 15;

    for (int t = tid; t < KS * 64; t += 128) kpad[t] = 0.0f;
    __syncthreads();
    for (int t = tid; t < KS * KS; t += 128) {
        int dy = t / KS, dx = t - dy * KS;
        float v = flipk ? kern[(KS - 1 - dy) * KS + (KS - 1 - dx)] : kern[t];
        kpad[dy * 64 + 16 + dx] = v;
    }

    unsigned tile = blockIdx.x * (unsigned)CWAVES + wave;
    int tx = tile & 127;
    int ty = (tile >> 7) & 127;
    int c  = tile >> 14;
    const float* inc = in + (size_t)c * (IMH * (size_t)IMW);
    int y0 = ty * 16, x0 = tx * 16;

    // Stage 40x40 wrap-padded patch (1600 = 50*32 -> no divergence)
    float* p = patch[wave];
    for (int t = lane; t < PATCH * PATCH; t += 32) {
        int r  = t / PATCH, cc = t - r * PATCH;
        int gy = (y0 + r  - KR) & (IMH - 1);
        int gx = (x0 + cc - KR) & (IMW - 1);
        p[r * PSTR + cc] = inc[(size_t)gy * IMW + gx];
    }
    __syncthreads();

    v8f acc = {};
    for (int dy = 0; dy < KS; ++dy) {
        const float* arow = p + (dy + l15) * PSTR;          // row dy+m of patch
        const float* krow = kpad + dy * 64 + 16 - l15;      // kern[dy, kk-n] = krow[kk]
#pragma unroll
        for (int kc = 0; kc < 10; ++kc) {
            int kk = kc * 4 + half * 2;
            v2f a = *(const v2f*)(arow + kk);               // A(m,kk), A(m,kk+1)  (b64)
            v2f b;
            b.x = krow[kk];                                 // B(kk , n)
            b.y = krow[kk + 1];                             // B(kk+1, n)
            acc = __builtin_amdgcn_wmma_f32_16x16x4_f32(
                false, a, false, b, (short)0, acc, false, false);
        }
    }

    float* oc = out + (size_t)c * (IMH * (size_t)IMW);
    int ox = x0 + l15;
#pragma unroll
    for (int r = 0; r < 8; ++r) {
        int oy = y0 + half * 8 + r;                         // C/D layout: VGPR r -> M=r / r+8
        oc[(size_t)oy * IMW + ox] = acc[r];
    }
}

// ---------------------------------------------------------------------------
// MATLAB imresize bicubic weights, computed on device in double (matches the
// numpy float64 reference). Down: 2048->512 (18 taps). Up: 512->2048 (6 taps).
// ---------------------------------------------------------------------------
__device__ inline double cubic_w(double x) {
    double ax = fabs(x), ax2 = ax * ax, ax3 = ax2 * ax;
    if (ax <= 1.0) return 1.5 * ax3 - 2.5 * ax2 + 1.0;
    if (ax <= 2.0) return -0.5 * ax3 + 2.5 * ax2 - 4.0 * ax + 2.0;
    return 0.0;
}

__global__ void gen_weights(float* __restrict__ wD, int* __restrict__ iD,
                            float* __restrict__ wU, int* __restrict__ iU)
{
    int t = blockIdx.x * blockDim.x + threadIdx.x;
    if (t < 512) {                       // downsample row t, scale = 0.25, kw = 16, P = 18
        int o = t;
        double u = 4.0 * (o + 1) - 1.5;
        double left = floor(u - 8.0);
        double w[18]; double s = 0.0;
        for (int q = 0; q < 18; ++q) { double ind = left + q; w[q] = 0.25 * cubic_w(0.25 * (u - ind)); s += w[q]; }
        for (int q = 0; q < 18; ++q) {
            long ind = (long)left + q;
            long m = ((ind - 1) % 4096 + 4096) % 4096;      // mirror over aux[4096]
            iD[o * 18 + q] = (m < 2048) ? (int)m : (int)(4095 - m);
            wD[o * 18 + q] = (float)(w[q] / s);
        }
    } else if (t < 512 + 2048) {         // upsample row, scale = 4, kw = 4, P = 6
        int o = t - 512;
        double u = 0.25 * (o + 1) + 0.375;
        double left = floor(u - 2.0);
        double w[6]; double s = 0.0;
        for (int q = 0; q < 6; ++q) { double ind = left + q; w[q] = cubic_w(u - ind); s += w[q]; }
        for (int q = 0; q < 6; ++q) {
            long ind = (long)left + q;
            long m = ((ind - 1) % 1024 + 1024) % 1024;
            iU[o * 6 + q] = (m < 512) ? (int)m : (int)(1023 - m);
            wU[o * 6 + q] = (float)(w[q] / s);
        }
    }
}

// t1[c,o,j] = sum_q wD[o,q] * x[c, iD[o,q], j]      (3,512,2048)
__global__ void down_rows(const float* __restrict__ x, const float* __restrict__ wD,
                          const int* __restrict__ iD, float* __restrict__ t1)
{
    int t = blockIdx.x * blockDim.x + threadIdx.x;
    if (t >= NCH * 512 * 2048) return;
    int j = t & 2047, o = (t >> 11) & 511, c = t >> 20;
    const float* xc = x + (size_t)c * (IMH * (size_t)IMW);
    float s = 0.f;
    for (int q = 0; q < 18; ++q) s += wD[o * 18 + q] * xc[(size_t)iD[o * 18 + q] * IMW + j];
    t1[t] = s;
}

// hx[c,o,p] = sum_q wD[p,q] * t1[c,o,iD[p,q]]       (3,512,512)
__global__ void down_cols(const float* __restrict__ t1, const float* __restrict__ wD,
                          const int* __restrict__ iD, float* __restrict__ hx)
{
    int t = blockIdx.x * blockDim.x + threadIdx.x;
    if (t >= NCH * 512 * 512) return;
    int p = t & 511, o = (t >> 9) & 511, c = t >> 18;
    const float* tr = t1 + ((size_t)c * 512 + o) * 2048;
    float s = 0.f;
    for (int q = 0; q < 18; ++q) s += wD[p * 18 + q] * tr[iD[p * 18 + q]];
    hx[t] = s;
}

// t2[c,O,p] = sum_q wU[O,q] * hx[c, iU[O,q], p]     (3,2048,512)
__global__ void up_rows(const float* __restrict__ hx, const float* __restrict__ wU,
                        const int* __restrict__ iU, float* __restrict__ t2)
{
    int t = blockIdx.x * blockDim.x + threadIdx.x;
    if (t >= NCH * 2048 * 512) return;
    int p = t & 511, O = (t >> 9) & 2047, c = t >> 20;
    const float* hc = hx + (size_t)c * 512 * 512;
    float s = 0.f;
    for (int q = 0; q < 6; ++q) s += wU[O * 6 + q] * hc[(size_t)iU[O * 6 + q] * 512 + p];
    t2[t] = s;
}

// out[c,O,P] = atax (already in out) + 0.1 * sum_q wU[P,q]*t2[c,O,iU[P,q]]
__global__ void up_cols_final(const float* __restrict__ t2, const float* __restrict__ wU,
                              const int* __restrict__ iU, float* __restrict__ out)
{
    int t = blockIdx.x * blockDim.x + threadIdx.x;
    if (t >= NCH * 2048 * 2048) return;
    int P = t & 2047, O = (t >> 11) & 2047, c = t >> 22;
    const float* tr = t2 + ((size_t)c * 2048 + O) * 512;
    float s = 0.f;
    for (int q = 0; q < 6; ++q) s += wU[P * 6 + q] * tr[iU[P * 6 + q]];
    out[t] = out[t] + 0.1f * s;
}

extern "C" void kernel_launch(void* const* d_in, const int* in_sizes, int n_in,
                              void* d_out, int out_size, void* d_ws, size_t ws_size,
                              hipStream_t stream)
{
    const float* im   = (const float*)d_in[0];   // (1,3,2048,2048) f32
    const float* kern = (const float*)d_in[1];   // (25,25) f32
    float* out = (float*)d_out;                  // (1,3,2048,2048) f32

    char* ws = (char*)d_ws;
    size_t off = 0;
    auto alloc = [&](size_t bytes) { void* r = ws + off; off += (bytes + 255) & ~(size_t)255; return r; };
    float* ax = (float*)alloc((size_t)NCH * IMH * IMW * 4);  // 50.3 MB
    float* t1 = (float*)alloc((size_t)NCH * 512 * 2048 * 4); // 12.6 MB
    float* hx = (float*)alloc((size_t)NCH * 512 * 512 * 4);  //  3.1 MB
    float* t2 = (float*)alloc((size_t)NCH * 2048 * 512 * 4); // 12.6 MB
    float* wD = (float*)alloc(512 * 18 * 4);
    int*   iD = (int*)  alloc(512 * 18 * 4);
    float* wU = (float*)alloc(2048 * 6 * 4);
    int*   iU = (int*)  alloc(2048 * 6 * 4);

    gen_weights<<<(2560 + 127) / 128, 128, 0, stream>>>(wD, iD, wU, iU);

    const int conv_blocks = (NCH * 128 * 128) / CWAVES;      // 12288
    conv25_wmma<<<conv_blocks, 128, 0, stream>>>(im, kern, ax,  1); // corr with flip(k)
    conv25_wmma<<<conv_blocks, 128, 0, stream>>>(ax, kern, out, 0); // corr with k -> atax

    int n;
    n = NCH * 512 * 2048;  down_rows   <<<(n + 255) / 256, 256, 0, stream>>>(im, wD, iD, t1);
    n = NCH * 512 * 512;   down_cols   <<<(n + 255) / 256, 256, 0, stream>>>(t1, wD, iD, hx);
    n = NCH * 2048 * 512;  up_rows     <<<(n + 255) / 256, 256, 0, stream>>>(hx, wU, iU, t2);
    n = NCH * 2048 * 2048; up_cols_final<<<(n + 255) / 256, 256, 0, stream>>>(t2, wU, iU, out);
}